// CANLayer_72645076844574
// MI455X (gfx1250) — compile-verified
//
#include <hip/hip_runtime.h>
#include <math.h>

#define N_CELLS 50000
#define N_EDGES 1600000
#define FDIM    64
#define EPS_SKIP (1.0f + 1e-6f)

typedef float v2f __attribute__((ext_vector_type(2)));
typedef float v8f __attribute__((ext_vector_type(8)));

__device__ __forceinline__ unsigned f32_order_key(float f) {
    unsigned u = __float_as_uint(f);
    return (u & 0x80000000u) ? ~u : (u | 0x80000000u);
}
__device__ __forceinline__ float f32_order_unkey(unsigned k) {
    return (k & 0x80000000u) ? __uint_as_float(k ^ 0x80000000u)
                             : __uint_as_float(~k);
}

// -------- out[r][c] = scale * sum_k X[r][k] * W[k][c];  X:[nrows,64] W:[64,64]
// One wave computes a 16x16 tile via 16x V_WMMA_F32_16X16X4_F32.
__global__ void __launch_bounds__(256) gemm_x_w64(const float* __restrict__ X,
                                                  const float* __restrict__ W,
                                                  float* __restrict__ out,
                                                  int nrow_tiles, float scale) {
    __shared__ float Wt[64 * 66];  // W transposed (Wt[c][k]), padded stride 66
    for (int i = threadIdx.x; i < 64 * 64; i += 256) {
        int r = i >> 6, c = i & 63;
        Wt[c * 66 + r] = W[i];
    }
    __syncthreads();

    const int wave = threadIdx.x >> 5;
    const int lane = threadIdx.x & 31;
    const int lo = lane & 15;      // M (A) / N (B) index
    const int hi = lane >> 4;      // K-pair selector
    const int row_tile = blockIdx.x * 2 + (wave >> 2);
    if (row_tile >= nrow_tiles) return;        // wave-uniform: EXEC stays all-1s
    const int row0 = row_tile * 16;
    const int col0 = (wave & 3) * 16;

    v8f acc = {};
    const float* xrow = X + (size_t)(row0 + lo) * 64 + 2 * hi;
    const float* wcol = Wt + (col0 + lo) * 66 + 2 * hi;
#pragma unroll
    for (int k = 0; k < 64; k += 4) {
        v2f a = *(const v2f*)(xrow + k);   // X[row][k+2hi], X[row][k+2hi+1]
        v2f b = *(const v2f*)(wcol + k);   // W[k+2hi][col], W[k+2hi+1][col]
        acc = __builtin_amdgcn_wmma_f32_16x16x4_f32(false, a, false, b,
                                                    0, acc, false, false);
    }
#pragma unroll
    for (int r = 0; r < 8; ++r)
        out[(size_t)(row0 + r + 8 * hi) * 64 + col0 + lo] = acc[r] * scale;
}

// -------- a_src[i] = xm[i] . att[0:64], a_tgt[i] = xm[i] . att[64:128]
__global__ void __launch_bounds__(256) att_dot(const float* __restrict__ xm,
                                               const float* __restrict__ att,
                                               float* __restrict__ a_src,
                                               float* __restrict__ a_tgt,
                                               int nrows) {
    int gid = blockIdx.x * blockDim.x + threadIdx.x;
    int wid = gid >> 5, lane = gid & 31;
    int nw = (gridDim.x * blockDim.x) >> 5;
    v2f cs = *(const v2f*)(att + 2 * lane);
    v2f ct = *(const v2f*)(att + 64 + 2 * lane);
    for (int row = wid; row < nrows; row += nw) {
        v2f v = *(const v2f*)(xm + (size_t)row * 64 + 2 * lane);
        float ps = v.x * cs.x + v.y * cs.y;
        float pt = v.x * ct.x + v.y * ct.y;
#pragma unroll
        for (int off = 16; off > 0; off >>= 1) {
            ps += __shfl_xor(ps, off, 32);
            pt += __shfl_xor(pt, off, 32);
        }
        if (lane == 0) { a_src[row] = ps; a_tgt[row] = pt; }
    }
}

// -------- pass1: e = elu(a_src[src]+a_tgt[tgt])*vals; segment-max via ordered-u32 atomicMax
__global__ void __launch_bounds__(256) edge_pass1(const int* __restrict__ tgt,
                                                  const int* __restrict__ src,
                                                  const float* __restrict__ vals,
                                                  const float* __restrict__ a_src,
                                                  const float* __restrict__ a_tgt,
                                                  float* __restrict__ ebuf,
                                                  unsigned* __restrict__ mord, int ne) {
    int e = blockIdx.x * blockDim.x + threadIdx.x;
    int stride = gridDim.x * blockDim.x;
    for (; e < ne; e += stride) {
        int t = tgt[e], s = src[e];
        float x = a_src[s] + a_tgt[t];
        float el = x > 0.f ? x : (__expf(x) - 1.f);   // elu
        float ev = el * vals[e];
        ebuf[e] = ev;
        atomicMax(&mord[t], f32_order_key(ev));
    }
}

// -------- pass2: z = exp(e - m[tgt]); denom[tgt] += z (overwrite ebuf with z)
__global__ void __launch_bounds__(256) edge_pass2(const int* __restrict__ tgt,
                                                  float* __restrict__ ebuf,
                                                  const unsigned* __restrict__ mord,
                                                  float* __restrict__ denom, int ne) {
    int e = blockIdx.x * blockDim.x + threadIdx.x;
    int stride = gridDim.x * blockDim.x;
    for (; e < ne; e += stride) {
        int t = tgt[e];
        float z = __expf(ebuf[e] - f32_order_unkey(mord[t]));
        ebuf[e] = z;
        unsafeAtomicAdd(&denom[t], z);
    }
}

// -------- pass3: acc[tgt] += (z/denom[tgt]) * xm[src]; one wave per edge
__global__ void __launch_bounds__(256) edge_pass3(const int* __restrict__ tgt,
                                                  const int* __restrict__ src,
                                                  const float* __restrict__ ebuf,
                                                  const float* __restrict__ denom,
                                                  const float* __restrict__ xm,
                                                  float* __restrict__ acc, int ne) {
    int gid = blockIdx.x * blockDim.x + threadIdx.x;
    int wid = gid >> 5, lane = gid & 31;
    int nw = (gridDim.x * blockDim.x) >> 5;
    for (int e = wid; e < ne; e += nw) {
        int t = tgt[e], s = src[e];
        float alpha = ebuf[e] / denom[t];
        v2f v = *(const v2f*)(xm + (size_t)s * 64 + 2 * lane);
        float* dst = acc + (size_t)t * 64 + 2 * lane;
        unsafeAtomicAdd(dst + 0, alpha * v.x);
        unsafeAtomicAdd(dst + 1, alpha * v.y);
    }
}

__global__ void __launch_bounds__(256) init_stats(unsigned* __restrict__ mord,
                                                  float* __restrict__ denom, int n) {
    int i = blockIdx.x * blockDim.x + threadIdx.x;
    if (i < n) { mord[i] = 0u; denom[i] = 0.f; }
}

__global__ void __launch_bounds__(256) relu_inplace(float* __restrict__ p, int n) {
    int i = blockIdx.x * blockDim.x + threadIdx.x;
    if (i < n) p[i] = fmaxf(p[i], 0.f);
}

extern "C" void kernel_launch(void* const* d_in, const int* in_sizes, int n_in,
                              void* d_out, int out_size, void* d_ws, size_t ws_size,
                              hipStream_t stream) {
    const float* x        = (const float*)d_in[0];
    const int*   lidx     = (const int*)d_in[1];   // [2,E]: row0 = tgt, row1 = src
    const float* lvals    = (const float*)d_in[2];
    const int*   uidx     = (const int*)d_in[3];
    const float* uvals    = (const float*)d_in[4];
    const float* w_lower  = (const float*)d_in[5];
    const float* att_low  = (const float*)d_in[6];
    const float* w_upper  = (const float*)d_in[7];
    const float* att_up   = (const float*)d_in[8];
    const float* w_lin    = (const float*)d_in[9];
    float* out = (float*)d_out;

    const int N = N_CELLS, E = N_EDGES;
    const size_t NF = (size_t)N * FDIM;

    // workspace layout (floats)
    float*    ws   = (float*)d_ws;
    float*    xmL  = ws;                       // N*64
    float*    xmU  = ws + NF;                  // N*64
    float*    aLs  = ws + 2 * NF;              // N
    float*    aLt  = aLs + N;                  // N
    float*    aUs  = aLt + N;                  // N
    float*    aUt  = aUs + N;                  // N
    unsigned* mord = (unsigned*)(aUt + N);     // 2*N (lower then upper)
    float*    den  = (float*)(mord + 2 * N);   // 2*N
    float*    eL   = den + 2 * N;              // E
    float*    eU   = eL + E;                   // E

    const int nrow_tiles = (N + 15) / 16;                 // 3125
    const int gemm_blocks = (nrow_tiles + 1) / 2;         // 2 row-tiles per block

    // 0) zero segment stats
    init_stats<<<(2 * N + 255) / 256, 256, 0, stream>>>(mord, den, 2 * N);

    // 1) three WMMA GEMMs; lin writes EPS * (x @ w_lin) straight into d_out (accumulator)
    gemm_x_w64<<<gemm_blocks, 256, 0, stream>>>(x, w_lower, xmL, nrow_tiles, 1.0f);
    gemm_x_w64<<<gemm_blocks, 256, 0, stream>>>(x, w_upper, xmU, nrow_tiles, 1.0f);
    gemm_x_w64<<<gemm_blocks, 256, 0, stream>>>(x, w_lin,  out, nrow_tiles, EPS_SKIP);

    // 2) attention dot products (wave per row)
    att_dot<<<6250, 256, 0, stream>>>(xmL, att_low, aLs, aLt, N);
    att_dot<<<6250, 256, 0, stream>>>(xmU, att_up,  aUs, aUt, N);

    const int eb = (E + 255) / 256;           // thread-per-edge blocks
    const int wb = (int)(((size_t)E * 32 + 255) / 256);  // wave-per-edge blocks

    // 3) lower conv
    edge_pass1<<<eb, 256, 0, stream>>>(lidx, lidx + E, lvals, aLs, aLt, eL, mord, E);
    edge_pass2<<<eb, 256, 0, stream>>>(lidx, eL, mord, den, E);
    edge_pass3<<<wb, 256, 0, stream>>>(lidx, lidx + E, eL, den, xmL, out, E);

    // 4) upper conv
    edge_pass1<<<eb, 256, 0, stream>>>(uidx, uidx + E, uvals, aUs, aUt, eU, mord + N, E);
    edge_pass2<<<eb, 256, 0, stream>>>(uidx, eU, mord + N, den + N, E);
    edge_pass3<<<wb, 256, 0, stream>>>(uidx, uidx + E, eU, den + N, xmU, out, E);

    // 5) final ReLU in place on the accumulator
    relu_inplace<<<(int)((NF + 255) / 256), 256, 0, stream>>>(out, (int)NF);
}